// AssoScan_85555748537031
// MI455X (gfx1250) — compile-verified
//
#include <hip/hip_runtime.h>
#include <cstdint>
#include <cstddef>

// Linear recurrence h_t = g_t*h_{t-1} + x_t, B=4, N=8192, D=1024, fp32.
// Memory-bound (0.17 FLOP/byte): 384 MB min traffic @ 23.3 TB/s ~= 16.5 us.
// 3-phase chunked scan; pass 3 streams via global_load_async_to_lds_b128
// (ASYNCcnt-paced ring buffer), inputs/out use NT hints so the 128 MB gates
// array stays resident in the 192 MB L2 between pass 1 and pass 3.

typedef float v4f __attribute__((ext_vector_type(4)));

constexpr int B_ = 4;
constexpr int N_ = 8192;
constexpr int D_ = 1024;
constexpr int S_ = 128;          // chunks along N
constexpr int L_ = N_ / S_;      // 64 steps per chunk
constexpr int T_ = 256;          // threads per block; each thread owns 4 channels
constexpr int PF = 7;            // async prefetch depth (steps)
constexpr size_t SEG = (size_t)B_ * S_ * D_;  // floats per ws segment

// ---------------------------------------------------------------------------
// Pass 1: per-chunk aggregates  A = prod(g), H = local inclusive scan end.
// Reads gates (RT -> stays in L2) and inputs (NT). Writes 4 MB of aggregates.
// ---------------------------------------------------------------------------
__global__ __launch_bounds__(T_) void scan_partials(const float* __restrict__ g,
                                                    const float* __restrict__ x,
                                                    float* __restrict__ wsA,
                                                    float* __restrict__ wsH) {
    const int bid = blockIdx.x;          // b*S_ + c
    const int c = bid % S_;
    const int b = bid / S_;
    const int t = threadIdx.x;
    const size_t base = ((size_t)b * N_ + (size_t)c * L_) * D_ + (size_t)t * 4;

    v4f A = {1.f, 1.f, 1.f, 1.f};
    v4f H = {0.f, 0.f, 0.f, 0.f};
#pragma unroll 4
    for (int s = 0; s < L_; ++s) {
        v4f gg = *(const v4f*)(g + base + (size_t)s * D_);
        v4f xx = __builtin_nontemporal_load((const v4f*)(x + base + (size_t)s * D_));
        H = gg * H + xx;
        A = A * gg;
    }
    const size_t w = (size_t)bid * D_ + (size_t)t * 4;
    *(v4f*)(wsA + w) = A;
    *(v4f*)(wsH + w) = H;
}

// ---------------------------------------------------------------------------
// Pass 2: serial combine of the S_ chunk aggregates per channel -> carry-in
// for every chunk.  carry[c] = H-state entering chunk c.  1024 threads total.
// ---------------------------------------------------------------------------
__global__ __launch_bounds__(T_) void scan_carries(const float* __restrict__ wsA,
                                                   const float* __restrict__ wsH,
                                                   float* __restrict__ wsC) {
    const int tid = blockIdx.x * T_ + threadIdx.x;   // 0 .. B_*D_/4-1
    const int b = tid / (D_ / 4);
    const int d = (tid % (D_ / 4)) * 4;
    v4f h = {0.f, 0.f, 0.f, 0.f};
    for (int c = 0; c < S_; ++c) {
        const size_t idx = ((size_t)(b * S_ + c)) * D_ + d;
        *(v4f*)(wsC + idx) = h;
        v4f A  = *(const v4f*)(wsA + idx);
        v4f Hl = *(const v4f*)(wsH + idx);
        h = A * h + Hl;
    }
}

// ---------------------------------------------------------------------------
// Pass 3: seeded local scan, streaming gates+inputs through LDS with
// global_load_async_to_lds_b128.  Per-wave ring of 8 slots (512B each per
// stream), depth-7 prefetch paced by s_wait_asynccnt.  gates re-read hits L2;
// inputs tagged TH_LOAD_NT; out stored non-temporally.
// ---------------------------------------------------------------------------
__global__ __launch_bounds__(T_) void scan_apply(const float* __restrict__ g,
                                                 const float* __restrict__ x,
                                                 const float* __restrict__ carry,
                                                 float* __restrict__ out) {
    // [wave][slot][lane] of v4f : 8*8*32 * 16B = 32 KB per stream.
    __shared__ v4f sg[8 * 8 * 32];
    __shared__ v4f sx[8 * 8 * 32];

    const int bid  = blockIdx.x;         // b*S_ + c
    const int c    = bid % S_;
    const int b    = bid / S_;
    const int t    = threadIdx.x;
    const int wave = t >> 5;
    const int lane = t & 31;

    const size_t rowBase = ((size_t)b * N_ + (size_t)c * L_) * D_;   // elem idx of step 0
    const unsigned long long gAddr = (unsigned long long)(size_t)(g + rowBase);
    const unsigned long long xAddr = (unsigned long long)(size_t)(x + rowBase);
    const unsigned voff = (unsigned)t * 16u;                          // lane byte offset in row

    // LDS byte offsets (low 32 bits of the generic address == LDS offset).
    const unsigned sgBase = (unsigned)(size_t)(void*)&sg[(wave * 8) * 32 + lane];
    const unsigned sxBase = (unsigned)(size_t)(void*)&sx[(wave * 8) * 32 + lane];

    auto issue = [&](int step) {
        const unsigned slot = (unsigned)step & 7u;
        const unsigned ldsg = sgBase + slot * 512u;
        const unsigned ldsx = sxBase + slot * 512u;
        const unsigned long long ga = gAddr + (unsigned long long)step * (D_ * 4);
        const unsigned long long xa = xAddr + (unsigned long long)step * (D_ * 4);
        // GVS form: lds_vgpr, offset_vgpr, sgpr64 base.  gates: RT (keep in L2).
        asm volatile("global_load_async_to_lds_b128 %0, %1, %2"
                     :: "v"(ldsg), "v"(voff), "s"(ga) : "memory");
        // inputs: non-temporal, don't evict gates from L2.
        asm volatile("global_load_async_to_lds_b128 %0, %1, %2 th:TH_LOAD_NT"
                     :: "v"(ldsx), "v"(voff), "s"(xa) : "memory");
    };

    for (int s = 0; s < PF; ++s) issue(s);

    v4f h = *(const v4f*)(carry + (size_t)bid * D_ + (size_t)t * 4);
    const size_t oIdx = rowBase + (size_t)t * 4;

    for (int s = 0; s < L_ - PF; ++s) {
        // issued = 2*(s+PF) instrs; outstanding <= 2*PF-2  =>  step s landed.
        asm volatile("s_wait_asynccnt 12" ::: "memory");
        const int slot = s & 7;
        v4f gg = sg[(wave * 8 + slot) * 32 + lane];
        v4f xx = sx[(wave * 8 + slot) * 32 + lane];
        h = gg * h + xx;
        __builtin_nontemporal_store(h, (v4f*)(out + oIdx + (size_t)s * D_));
        issue(s + PF);   // overwrites slot (s-1)&7, already consumed above
    }
    asm volatile("s_wait_asynccnt 0" ::: "memory");
    for (int s = L_ - PF; s < L_; ++s) {
        const int slot = s & 7;
        v4f gg = sg[(wave * 8 + slot) * 32 + lane];
        v4f xx = sx[(wave * 8 + slot) * 32 + lane];
        h = gg * h + xx;
        __builtin_nontemporal_store(h, (v4f*)(out + oIdx + (size_t)s * D_));
    }
}

// ---------------------------------------------------------------------------
extern "C" void kernel_launch(void* const* d_in, const int* in_sizes, int n_in,
                              void* d_out, int out_size, void* d_ws, size_t ws_size,
                              hipStream_t stream) {
    const float* gates  = (const float*)d_in[0];
    const float* inputs = (const float*)d_in[1];
    float* out = (float*)d_out;
    float* ws  = (float*)d_ws;            // needs 3*SEG*4 = 6 MB
    float* wsA = ws;
    float* wsH = ws + SEG;
    float* wsC = ws + 2 * SEG;

    scan_partials<<<B_ * S_, T_, 0, stream>>>(gates, inputs, wsA, wsH);
    scan_carries<<<(B_ * D_ / 4) / T_, T_, 0, stream>>>(wsA, wsH, wsC);
    scan_apply<<<B_ * S_, T_, 0, stream>>>(gates, inputs, wsC, out);
}